// SpatialTransformer_35742717837651
// MI455X (gfx1250) — compile-verified
//
#include <hip/hip_runtime.h>

// 3D trilinear warp (grid_sample, zeros padding), fp32.
// src:  [B=2, C=1, D=160, H=192, W=224]
// flow: [B=2, 3,    D,     H,     W   ]  (d,h,w displacements in voxels)
// out:  [B=2, C=1, D, H, W]
//
// Memory-bound (~275 MB HBM, ~12us floor @ 23.3 TB/s); no matmul structure so
// WMMA is inapplicable. Round-4: revert the round-3 regressions (back to the
// round-2 index/address form) and vectorize the fp32 pipeline as voxel-pairs
// (float2 ext-vectors) so the backend can use CDNA5 packed-f32 VOP3P ops
// (v_pk_add_f32 / v_pk_mul_f32 / v_pk_fma_f32) -- 2 fp ops per instruction
// for the ~60% of VALU work that is plain fp add/mul/fma.

constexpr int Dd = 160, Hh = 192, Ww = 224, Bb = 2;
constexpr int HW   = Hh * Ww;          // 43008
constexpr int Nvol = Dd * HW;          // 6,881,280
constexpr int VEC  = 4;                // voxels per thread (W % 4 == 0)
constexpr int THREADS = 256;           // 8 wave32s per block
constexpr int BLOCKS_PER_SLICE = HW / (VEC * THREADS);   // 42 exactly

typedef float v4f __attribute__((ext_vector_type(4)));
typedef float v2f __attribute__((ext_vector_type(2)));

__global__ __launch_bounds__(THREADS) void warp3d_kernel(
    const float* __restrict__ src, const float* __restrict__ flow,
    float* __restrict__ out)
{
    const int d  = blockIdx.y;                // scalar (wave-uniform)
    const int b  = blockIdx.z;                // scalar
    const int r2 = (blockIdx.x * THREADS + threadIdx.x) * VEC;  // 0..43004
    const int h  = r2 / Ww;                   // one small magic-divide
    const int w  = r2 - h * Ww;

    // Wave-uniform bases -> saddr-form global accesses with 32-bit voffsets.
    const float* __restrict__ srcb = src + (size_t)b * Nvol;
    const float* __restrict__ flb  = flow + (size_t)b * 3 * Nvol + (size_t)d * HW;
    float* __restrict__ outb       = out + (size_t)b * Nvol + (size_t)d * HW;

    // Warm the src lines at the identity position while the flow loads are in
    // flight (flow ~ N(0,1) voxels => corners land on/near these lines).
    // global_prefetch_b8: fire-and-forget, no LOADcnt.
    const float* p0 = srcb + (size_t)d * HW + r2;
    __builtin_prefetch(p0, 0, 3);
    __builtin_prefetch(p0 + Ww, 0, 3);
    __builtin_prefetch(p0 + HW, 0, 3);

    // Streamed read-once data: non-temporal so it does not evict src from L2.
    const v4f f_d = __builtin_nontemporal_load((const v4f*)(flb + r2));
    const v4f f_h = __builtin_nontemporal_load((const v4f*)(flb + Nvol + r2));
    const v4f f_w = __builtin_nontemporal_load((const v4f*)(flb + 2 * Nvol + r2));

    v4f od;
#pragma unroll
    for (int p = 0; p < 2; ++p) {
        const int i0 = 2 * p, i1 = 2 * p + 1;

        // ---- packed fp32 section (pairs -> v_pk_* candidates) ----
        const v2f fD = {f_d[i0], f_d[i1]};
        const v2f fH = {f_h[i0], f_h[i1]};
        const v2f fW = {f_w[i0], f_w[i1]};

        const v2f cd = (float)d + fD;
        const v2f ch = (float)h + fH;
        const v2f cw = v2f{(float)(w + i0), (float)(w + i1)} + fW;

        // floor is VOP1 (no packed form) -> scalar per element.
        const v2f d0f = {floorf(cd.x), floorf(cd.y)};
        const v2f h0f = {floorf(ch.x), floorf(ch.y)};
        const v2f w0f = {floorf(cw.x), floorf(cw.y)};

        const v2f frD = cd - d0f;            // pk_add(neg)
        const v2f frH = ch - h0f;
        const v2f frW = cw - w0f;
        const v2f ofD = 1.0f - frD;
        const v2f ofH = 1.0f - frH;
        const v2f ofW = 1.0f - frW;

        // ---- per-element integer / select section ----
        const int d0e[2] = {(int)d0f.x, (int)d0f.y};
        const int h0e[2] = {(int)h0f.x, (int)h0f.y};
        const int w0e[2] = {(int)w0f.x, (int)w0f.y};

        v2f wd0, wd1, wh0, wh1, ww0, ww1;
        v2f v000, v001, v010, v011, v100, v101, v110, v111;
#pragma unroll
        for (int e = 0; e < 2; ++e) {
            const int d0 = d0e[e], h0 = h0e[e], w0 = w0e[e];
            const int d1 = d0 + 1, h1 = h0 + 1, w1 = w0 + 1;

            // Zeros padding: per-axis weight zeroed when out of range;
            // (unsigned)x < N is one compare per bound.
            wd0[e] = ((unsigned)d0 < (unsigned)Dd) ? ofD[e] : 0.f;
            wd1[e] = ((unsigned)d1 < (unsigned)Dd) ? frD[e] : 0.f;
            wh0[e] = ((unsigned)h0 < (unsigned)Hh) ? ofH[e] : 0.f;
            wh1[e] = ((unsigned)h1 < (unsigned)Hh) ? frH[e] : 0.f;
            ww0[e] = ((unsigned)w0 < (unsigned)Ww) ? ofW[e] : 0.f;
            ww1[e] = ((unsigned)w1 < (unsigned)Ww) ? frW[e] : 0.f;

            // Clamped indices: all 8 gathers unconditional & in-bounds; the
            // zeroed weights kill out-of-range contributions (no divergence).
            const int d0c = min(max(d0, 0), Dd - 1), d1c = min(max(d1, 0), Dd - 1);
            const int h0c = min(max(h0, 0), Hh - 1), h1c = min(max(h1, 0), Hh - 1);
            const int w0c = min(max(w0, 0), Ww - 1), w1c = min(max(w1, 0), Ww - 1);

            const int rowA = d0c * HW, rowB = d1c * HW;
            const int colA = h0c * Ww, colB = h1c * Ww;
            const int pAA = rowA + colA, pAB = rowA + colB;
            const int pBA = rowB + colA, pBB = rowB + colB;

            v000[e] = srcb[pAA + w0c];
            v001[e] = srcb[pAA + w1c];
            v010[e] = srcb[pAB + w0c];
            v011[e] = srcb[pAB + w1c];
            v100[e] = srcb[pBA + w0c];
            v101[e] = srcb[pBA + w1c];
            v110[e] = srcb[pBB + w0c];
            v111[e] = srcb[pBB + w1c];
        }

        // ---- packed fp32 blend (v_pk_mul / v_pk_fma candidates) ----
        const v2f q00 = wh0 * ww0, q01 = wh0 * ww1;
        const v2f q10 = wh1 * ww0, q11 = wh1 * ww1;

        v2f t0 = v000 * q00;
        t0 = __builtin_elementwise_fma(v001, q01, t0);
        t0 = __builtin_elementwise_fma(v010, q10, t0);
        t0 = __builtin_elementwise_fma(v011, q11, t0);
        v2f t1 = v100 * q00;
        t1 = __builtin_elementwise_fma(v101, q01, t1);
        t1 = __builtin_elementwise_fma(v110, q10, t1);
        t1 = __builtin_elementwise_fma(v111, q11, t1);

        const v2f res = __builtin_elementwise_fma(t1, wd1, t0 * wd0);
        od[i0] = res.x;
        od[i1] = res.y;
    }

    // Write-once stream: non-temporal store, keeps L2 for src.
    __builtin_nontemporal_store(od, (v4f*)(outb + r2));
}

extern "C" void kernel_launch(void* const* d_in, const int* in_sizes, int n_in,
                              void* d_out, int out_size, void* d_ws, size_t ws_size,
                              hipStream_t stream) {
    const float* src  = (const float*)d_in[0];   // [2,1,160,192,224] fp32
    const float* flow = (const float*)d_in[1];   // [2,3,160,192,224] fp32
    float* out = (float*)d_out;                  // [2,1,160,192,224] fp32

    dim3 grid(BLOCKS_PER_SLICE, Dd, Bb);         // (42, 160, 2), exact cover
    warp3d_kernel<<<grid, THREADS, 0, stream>>>(src, flow, out);
}